// GCNModel_60086592471688
// MI455X (gfx1250) — compile-verified
//
#include <hip/hip_runtime.h>
#include <hip/hip_bf16.h>
#include <stdint.h>

// ---------------------------------------------------------------------------
// GCNConv forward for MI455X (gfx1250, wave32, WMMA).
//   xp  = x @ W^T                        (fp32 WMMA 16x16x4)
//   deg = 1 + scatter-count(dst)         (f32 atomics, L2 resident)
//   out = xp*dinv^2 + b  (self loop)     then  out[dst] += xp[src]*dinv_s*dinv_d
//   out = log_softmax(out, axis=1)       (wave-per-node, width-32 shuffles)
// ---------------------------------------------------------------------------

#define D_FEAT 256
#define N_CLS  64

typedef __attribute__((ext_vector_type(2))) float v2f;
typedef __attribute__((ext_vector_type(8))) float v8f;

// ---------------- GEMM: xp = x @ W^T via V_WMMA_F32_16X16X4_F32 -------------
// One wave computes a 16(node-rows) x 64(classes) tile: 4 accumulators, K=256.
__global__ __launch_bounds__(256) void gcn_gemm_wmma(
    const float* __restrict__ x, const float* __restrict__ W,
    float* __restrict__ xp, int n_nodes) {
  const int lane = threadIdx.x & 31;
  const int wave = threadIdx.x >> 5;
  const int tile = blockIdx.x * 8 + wave;        // 16-row tile (wave-uniform)
  if (tile * 16 >= n_nodes) return;              // uniform guard: EXEC all-1s
  const int row0 = tile * 16;
  const int half = lane >> 4;                    // 0: K pair {0,1}; 1: {2,3}
  const int l15  = lane & 15;

  const float* xrow = x + (size_t)(row0 + l15) * D_FEAT;
  v8f acc0 = {}, acc1 = {}, acc2 = {}, acc3 = {};

#pragma unroll 4
  for (int k0 = 0; k0 < D_FEAT; k0 += 4) {
    // A: 16x4 fp32 tile of x. lane l15 = row, half selects K sub-pair.
    v2f a = *(const v2f*)(xrow + k0 + 2 * half);
    // B: 4x16 tiles of W^T; B[k][n] = W[n][k]. l15 = column n within tile.
    v2f b0 = *(const v2f*)(W + (size_t)(0  + l15) * D_FEAT + k0 + 2 * half);
    v2f b1 = *(const v2f*)(W + (size_t)(16 + l15) * D_FEAT + k0 + 2 * half);
    v2f b2 = *(const v2f*)(W + (size_t)(32 + l15) * D_FEAT + k0 + 2 * half);
    v2f b3 = *(const v2f*)(W + (size_t)(48 + l15) * D_FEAT + k0 + 2 * half);
    acc0 = __builtin_amdgcn_wmma_f32_16x16x4_f32(false, a, false, b0, (short)0, acc0, false, false);
    acc1 = __builtin_amdgcn_wmma_f32_16x16x4_f32(false, a, false, b1, (short)0, acc1, false, false);
    acc2 = __builtin_amdgcn_wmma_f32_16x16x4_f32(false, a, false, b2, (short)0, acc2, false, false);
    acc3 = __builtin_amdgcn_wmma_f32_16x16x4_f32(false, a, false, b3, (short)0, acc3, false, false);
  }

  // C/D layout: VGPR r holds row (r + 8*half), column = l15 within N-tile.
#pragma unroll
  for (int r = 0; r < 8; ++r) {
    const int m = r + half * 8;
    float* orow = xp + (size_t)(row0 + m) * N_CLS + l15;
    orow[0]  = acc0[r];
    orow[16] = acc1[r];
    orow[32] = acc2[r];
    orow[48] = acc3[r];
  }
}

// ---------------- degree / normalization ------------------------------------
__global__ void gcn_deg_init(float* __restrict__ deg, int n) {
  int i = blockIdx.x * blockDim.x + threadIdx.x;
  if (i < n) deg[i] = 1.0f;  // self loop
}

__global__ void gcn_deg_accum(const int64_t* __restrict__ dst,
                              float* __restrict__ deg, int n_edges) {
  int e = blockIdx.x * blockDim.x + threadIdx.x;
  if (e < n_edges) atomicAdd(&deg[(int)dst[e]], 1.0f);
}

__global__ void gcn_dinv(float* __restrict__ deg, int n) {
  int i = blockIdx.x * blockDim.x + threadIdx.x;
  if (i < n) {
    float d = deg[i];
    deg[i] = (d > 0.0f) ? rsqrtf(d) : 0.0f;  // deg -> dinv in place
  }
}

// out = self-loop contribution + bias  (seeds the aggregation buffer)
__global__ void gcn_out_init(const float* __restrict__ xp,
                             const float* __restrict__ dinv,
                             const float* __restrict__ b,
                             float* __restrict__ out, int n_nodes) {
  int idx = blockIdx.x * blockDim.x + threadIdx.x;
  if (idx < n_nodes * N_CLS) {
    int i = idx >> 6, c = idx & 63;
    float di = dinv[i];
    out[idx] = xp[idx] * di * di + b[c];
  }
}

// ---------------- edge scatter: one wave per edge, 2 feats/lane -------------
__global__ __launch_bounds__(256) void gcn_scatter(
    const float* __restrict__ xp, const float* __restrict__ dinv,
    const int64_t* __restrict__ src, const int64_t* __restrict__ dst,
    float* __restrict__ out, int n_edges) {
  int e = blockIdx.x * 8 + (threadIdx.x >> 5);
  if (e >= n_edges) return;
  const int lane = threadIdx.x & 31;
  const int s = (int)src[e];
  const int d = (int)dst[e];
  const float norm = dinv[s] * dinv[d];
  const float* xs = xp + (size_t)s * N_CLS;
  float* od = out + (size_t)d * N_CLS;
  atomicAdd(&od[lane],      xs[lane]      * norm);
  atomicAdd(&od[lane + 32], xs[lane + 32] * norm);
}

// ---------------- log_softmax over 64 classes: one wave per node ------------
__global__ __launch_bounds__(256) void gcn_logsoftmax(float* __restrict__ out,
                                                      int n_nodes) {
  int node = blockIdx.x * 8 + (threadIdx.x >> 5);
  if (node >= n_nodes) return;
  const int lane = threadIdx.x & 31;
  float* row = out + (size_t)node * N_CLS;
  float v0 = row[lane], v1 = row[lane + 32];

  float m = fmaxf(v0, v1);
#pragma unroll
  for (int off = 16; off >= 1; off >>= 1)
    m = fmaxf(m, __shfl_xor(m, off, 32));   // wave32: explicit width 32

  float s = __expf(v0 - m) + __expf(v1 - m);
#pragma unroll
  for (int off = 16; off >= 1; off >>= 1)
    s += __shfl_xor(s, off, 32);

  const float lse = m + __logf(s);
  row[lane]      = v0 - lse;
  row[lane + 32] = v1 - lse;
}

// ---------------------------------------------------------------------------
extern "C" void kernel_launch(void* const* d_in, const int* in_sizes, int n_in,
                              void* d_out, int out_size, void* d_ws, size_t ws_size,
                              hipStream_t stream) {
  const float*   x  = (const float*)d_in[0];
  const int64_t* ei = (const int64_t*)d_in[1];   // [2, E] int64
  const float*   W  = (const float*)d_in[2];     // [64, 256]
  const float*   b  = (const float*)d_in[3];     // [64]
  float*         out = (float*)d_out;            // [N, 64]

  const int n_nodes = in_sizes[0] / D_FEAT;
  const int n_edges = in_sizes[1] / 2;
  const int64_t* src = ei;
  const int64_t* dst = ei + n_edges;

  // workspace: xp [N*64] f32, then dinv [N] f32
  float* xp   = (float*)d_ws;
  float* dinv = xp + (size_t)n_nodes * N_CLS;

  // 1) GEMM via fp32 WMMA: 8 waves/block, 16 rows per wave.
  {
    int tiles  = (n_nodes + 15) / 16;
    int blocks = (tiles + 7) / 8;
    gcn_gemm_wmma<<<blocks, 256, 0, stream>>>(x, W, xp, n_nodes);
  }
  // 2) degree (with self loops) -> dinv
  gcn_deg_init<<<(n_nodes + 255) / 256, 256, 0, stream>>>(dinv, n_nodes);
  gcn_deg_accum<<<(n_edges + 255) / 256, 256, 0, stream>>>(dst, dinv, n_edges);
  gcn_dinv<<<(n_nodes + 255) / 256, 256, 0, stream>>>(dinv, n_nodes);
  // 3) seed out with self-loop term + bias
  gcn_out_init<<<((size_t)n_nodes * N_CLS + 255) / 256, 256, 0, stream>>>(
      xp, dinv, b, out, n_nodes);
  // 4) edge scatter-add (one wave per edge)
  gcn_scatter<<<(n_edges + 7) / 8, 256, 0, stream>>>(xp, dinv, src, dst, out,
                                                     n_edges);
  // 5) log_softmax in place
  gcn_logsoftmax<<<(n_nodes + 7) / 8, 256, 0, stream>>>(out, n_nodes);
}